// AttentionAggregationModule_33638183862625
// MI455X (gfx1250) — compile-verified
//
#include <hip/hip_runtime.h>
#include <math.h>

typedef float v2f __attribute__((ext_vector_type(2)));
typedef float v8f __attribute__((ext_vector_type(8)));

#define NCOLS 65536   // total columns = B * H * W
#define NPB   16384   // columns per batch (H*W)

// ---------------------------------------------------------------------------
// Generic f32 WMMA GEMM:  C[m][col] = sum_k A[b][m][k] * B[k][col] (+bias[b][m])
// mode 0: B element = Bp[k*NCOLS + col]
// mode 1: B element gathered from the virtual concat [s5;s4;s3;s2]
// Tile: 128(M) x 128(N) x 32(K), 8 waves; wave w owns 16-row strip w and all
// eight 16-col subtiles -> one A-fragment feeds 8 WMMAs (reuse-A friendly).
// LDS: A row-major pad-34 (even) and B transposed pad-34 so both fragment
// element pairs are contiguous 8B-aligned -> single ds_load_b64 each.
// ---------------------------------------------------------------------------
__global__ __launch_bounds__(256)
void gemm_f32_wmma(const float* __restrict__ A, int M, int K, int aBatchStride,
                   const float* __restrict__ Bp,
                   const float* __restrict__ g0, const float* __restrict__ g1,
                   const float* __restrict__ g2, const float* __restrict__ g3,
                   int mode,
                   const float* __restrict__ bias, int biasBatchStride,
                   float* __restrict__ Cout)
{
    __shared__ float As[128][34];    // [m][k]
    __shared__ float Bst[128][34];   // [n][k]  (transposed)

    const int t    = threadIdx.x;
    const int lane = t & 31;
    const int wave = t >> 5;         // m-strip 0..7
    const int n0   = blockIdx.x * 128;
    const int m0   = blockIdx.y * 128;
    const int b    = n0 / NPB;       // 128 | 16384 -> whole tile in one batch

    v8f acc[8];
#pragma unroll
    for (int j = 0; j < 8; ++j)
#pragma unroll
        for (int i = 0; i < 8; ++i) acc[j][i] = 0.0f;

    const float* Ab = A + (size_t)b * aBatchStride;
    const int nloc = n0 & (NPB - 1);

    for (int k0 = 0; k0 < K; k0 += 32) {
        // ---- load A tile 128x32 (16 elems/thread, coalesced 32-elem rows) ----
#pragma unroll
        for (int i = 0; i < 16; ++i) {
            int idx = i * 256 + t;
            int m = idx >> 5, k = idx & 31;
            int gm = m0 + m;
            As[m][k] = (gm < M) ? Ab[(size_t)gm * K + k0 + k] : 0.0f;
        }
        // ---- load B tile 32x128, store transposed (coalesced global reads) ----
        if (mode == 0) {
#pragma unroll
            for (int i = 0; i < 16; ++i) {
                int idx = i * 256 + t;
                int k = idx >> 7, n = idx & 127;
                Bst[n][k] = Bp[(size_t)(k0 + k) * NCOLS + n0 + n];
            }
        } else {
#pragma unroll
            for (int i = 0; i < 16; ++i) {
                int idx = i * 256 + t;
                int k = idx >> 7, n = idx & 127;
                int cin = k0 + k;
                const float* src = (cin < 256) ? g0 : (cin < 512) ? g1
                                  : (cin < 768) ? g2 : g3;
                Bst[n][k] = src[((size_t)(b * 256 + (cin & 255))) * NPB + nloc + n];
            }
        }
        __syncthreads();

        // ---- 8 k-steps; 8 WMMAs per step sharing one A fragment ----
#pragma unroll
        for (int kk = 0; kk < 32; kk += 4) {
            const int kidx = kk + ((lane >> 4) << 1);   // even -> 8B aligned
            v2f a = *(const v2f*)&As[wave * 16 + (lane & 15)][kidx];
#pragma unroll
            for (int j = 0; j < 8; ++j) {
                v2f bf = *(const v2f*)&Bst[j * 16 + (lane & 15)][kidx];
                acc[j] = __builtin_amdgcn_wmma_f32_16x16x4_f32(
                    false, a, false, bf, (short)0, acc[j], false, false);
            }
        }
        __syncthreads();
    }

    // ---- epilogue: optional bias, guarded store ----
#pragma unroll
    for (int j = 0; j < 8; ++j) {
        int col = n0 + j * 16 + (lane & 15);
        const float* accp = (const float*)&acc[j];
#pragma unroll
        for (int i = 0; i < 8; ++i) {
            int gm = m0 + wave * 16 + i + ((lane >> 4) << 3);
            if (gm < M) {
                float v = accp[i];
                if (bias) v += bias[b * biasBatchStride + gm];
                Cout[(size_t)gm * NCOLS + col] = v;
            }
        }
    }
}

// ---------------------------------------------------------------------------
// BatchNorm stats: one block per channel, fp64 accumulation (deterministic).
// ---------------------------------------------------------------------------
__global__ __launch_bounds__(256)
void bn_stats(const float* __restrict__ x, const float* __restrict__ gamma,
              const float* __restrict__ beta,
              float* __restrict__ scale, float* __restrict__ shift)
{
    __shared__ double s1[256], s2[256];
    const int c = blockIdx.x, t = threadIdx.x;
    const float* row = x + (size_t)c * NCOLS;
    double a = 0.0, q = 0.0;
    for (int i = t; i < NCOLS; i += 256) { double v = row[i]; a += v; q += v * v; }
    s1[t] = a; s2[t] = q; __syncthreads();
    for (int s = 128; s > 0; s >>= 1) {
        if (t < s) { s1[t] += s1[t + s]; s2[t] += s2[t + s]; }
        __syncthreads();
    }
    if (t == 0) {
        double mu  = s1[0] / (double)NCOLS;
        double var = s2[0] / (double)NCOLS - mu * mu;
        float r = (float)(1.0 / sqrt(var + 1e-5));
        float g = gamma[c];
        scale[c] = g * r;
        shift[c] = beta[c] - (float)(mu) * g * r;
    }
}

__global__ __launch_bounds__(256)
void bn_relu_inplace(float* __restrict__ x, const float* __restrict__ scale,
                     const float* __restrict__ shift)
{
    size_t i4 = ((size_t)blockIdx.x * 256 + threadIdx.x) * 4;
    int c = (int)(i4 >> 16);
    float s = scale[c], sh = shift[c];
    float4 v = *(float4*)(x + i4);
    v.x = fmaxf(fmaf(v.x, s, sh), 0.0f);
    v.y = fmaxf(fmaf(v.y, s, sh), 0.0f);
    v.z = fmaxf(fmaf(v.z, s, sh), 0.0f);
    v.w = fmaxf(fmaf(v.w, s, sh), 0.0f);
    *(float4*)(x + i4) = v;
}

// ---------------------------------------------------------------------------
// Per-column L2 normalization of Q and K over the 32 channels (in place).
// ---------------------------------------------------------------------------
__global__ __launch_bounds__(256)
void l2norm_inplace(float* __restrict__ Q, float* __restrict__ K)
{
    const int col = blockIdx.x * 256 + threadIdx.x;
    float sq = 0.0f, sk = 0.0f;
#pragma unroll 8
    for (int r = 0; r < 32; ++r) {
        float q = Q[(size_t)r * NCOLS + col]; sq += q * q;
        float k = K[(size_t)r * NCOLS + col]; sk += k * k;
    }
    float iq = 1.0f / fmaxf(sqrtf(sq), 1e-6f);
    float ik = 1.0f / fmaxf(sqrtf(sk), 1e-6f);
#pragma unroll 8
    for (int r = 0; r < 32; ++r) {
        Q[(size_t)r * NCOLS + col] *= iq;
        K[(size_t)r * NCOLS + col] *= ik;
    }
}

// Row sums over one batch's 16384 columns. grid = B*rows blocks.
__global__ __launch_bounds__(256)
void rowsum16k(const float* __restrict__ src, float* __restrict__ dst, int rows)
{
    __shared__ float s[256];
    const int bid = blockIdx.x;
    const int b = bid / rows, r = bid - b * rows;
    const float* p = src + (size_t)r * NCOLS + (size_t)b * NPB;
    float a = 0.0f;
    for (int i = threadIdx.x; i < NPB; i += 256) a += p[i];
    s[threadIdx.x] = a; __syncthreads();
    for (int st = 128; st > 0; st >>= 1) {
        if (threadIdx.x < st) s[threadIdx.x] += s[threadIdx.x + st];
        __syncthreads();
    }
    if (threadIdx.x == 0) dst[b * rows + r] = s[0];
}

// tailor[col] = 1 / max(N + sum_q Qn[q][col]*Ksum[b][q], 1e-6)
__global__ __launch_bounds__(256)
void tailor_kernel(const float* __restrict__ Qn, const float* __restrict__ Ksum,
                   float* __restrict__ tl)
{
    const int col = blockIdx.x * 256 + threadIdx.x;
    const int b = col >> 14;
    float e = 0.0f;
#pragma unroll 8
    for (int q = 0; q < 32; ++q)
        e += Qn[(size_t)q * NCOLS + col] * Ksum[b * 32 + q];
    tl[col] = 1.0f / fmaxf((float)NPB + e, 1e-6f);
}

// ---------------------------------------------------------------------------
// split-K partials of matrix[b][q][c] = sum_n Kn[q][n]*V[c][n].
// grid (nsplit=8, ctile=8, b=4); deterministic (no atomics).
// ---------------------------------------------------------------------------
__global__ __launch_bounds__(256)
void kv_partial(const float* __restrict__ Kn, const float* __restrict__ V,
                float* __restrict__ part)
{
    __shared__ float Ks[32][65];
    __shared__ float Vs[32][65];
    const int ns = blockIdx.x, ct = blockIdx.y, b = blockIdx.z;
    const int t = threadIdx.x;
    const int q = t & 31;
    const int c0 = (t >> 5) * 4;
    float a0 = 0, a1 = 0, a2 = 0, a3 = 0;
    const size_t base = (size_t)b * NPB + (size_t)ns * 2048;
    for (int nb = 0; nb < 2048; nb += 64) {
#pragma unroll
        for (int i = 0; i < 8; ++i) {
            int idx = i * 256 + t;
            int row = idx >> 6, nn = idx & 63;
            Ks[row][nn] = Kn[(size_t)row * NCOLS + base + nb + nn];
            Vs[row][nn] = V[(size_t)(ct * 32 + row) * NCOLS + base + nb + nn];
        }
        __syncthreads();
#pragma unroll 4
        for (int nn = 0; nn < 64; ++nn) {
            float kq = Ks[q][nn];
            a0 = fmaf(kq, Vs[c0 + 0][nn], a0);
            a1 = fmaf(kq, Vs[c0 + 1][nn], a1);
            a2 = fmaf(kq, Vs[c0 + 2][nn], a2);
            a3 = fmaf(kq, Vs[c0 + 3][nn], a3);
        }
        __syncthreads();
    }
    size_t o = ((((size_t)ns * 4 + b) * 32 + q) * 256) + ct * 32 + c0;
    part[o + 0] = a0; part[o + 1] = a1; part[o + 2] = a2; part[o + 3] = a3;
}

// Reduce 8 partials and store transposed: matT[b][c][q]
__global__ __launch_bounds__(256)
void kv_reduceT(const float* __restrict__ part, float* __restrict__ matT)
{
    const int idx = blockIdx.x * 256 + threadIdx.x;  // 32768 total
    const int b = idx >> 13;
    const int r = idx & 8191;
    const int q = r >> 8;
    const int c = r & 255;
    float s = 0.0f;
#pragma unroll
    for (int ns = 0; ns < 8; ++ns)
        s += part[(((size_t)ns * 4 + b) * 32 + q) * 256 + c];
    matT[((size_t)b * 256 + c) * 32 + q] = s;
}

__device__ __forceinline__ float nan_fix(float r)
{
    if (__builtin_isnan(r)) return 0.0f;
    if (__builtin_isinf(r)) return r > 0.0f ? 1.0f : -1.0f;
    return r;
}

// out[b][c][n] = nan_fix(gamma * S[c][col] * tailor[col]) + feat[c][col]
__global__ __launch_bounds__(256)
void final_kernel(const float* __restrict__ S, const float* __restrict__ tl,
                  const float* __restrict__ feat, const float* __restrict__ gamma,
                  float* __restrict__ out)
{
    size_t i4 = ((size_t)blockIdx.x * 256 + threadIdx.x) * 4;
    const int b = (int)(i4 / ((size_t)256 * NPB));
    const int rem = (int)(i4 - (size_t)b * 256 * NPB);
    const int c = rem >> 14;
    const int n = rem & (NPB - 1);
    const int col = b * NPB + n;
    const float g = gamma[0];
    float4 sv = *(const float4*)(S + (size_t)c * NCOLS + col);
    float4 tv = *(const float4*)(tl + col);
    float4 fv = *(const float4*)(feat + (size_t)c * NCOLS + col);
    float4 o;
    o.x = nan_fix(g * sv.x * tv.x) + fv.x;
    o.y = nan_fix(g * sv.y * tv.y) + fv.y;
    o.z = nan_fix(g * sv.z * tv.z) + fv.z;
    o.w = nan_fix(g * sv.w * tv.w) + fv.w;
    *(float4*)(out + i4) = o;
}

// ---------------------------------------------------------------------------
extern "C" void kernel_launch(void* const* d_in, const int* in_sizes, int n_in,
                              void* d_out, int out_size, void* d_ws, size_t ws_size,
                              hipStream_t stream)
{
    (void)in_sizes; (void)n_in; (void)out_size; (void)ws_size;
    const float* s5      = (const float*)d_in[0];
    const float* s4      = (const float*)d_in[1];
    const float* s3      = (const float*)d_in[2];
    const float* s2      = (const float*)d_in[3];
    const float* w_conv  = (const float*)d_in[4];
    const float* bn_g    = (const float*)d_in[5];
    const float* bn_b    = (const float*)d_in[6];
    const float* wq      = (const float*)d_in[7];
    const float* bq      = (const float*)d_in[8];
    const float* wk      = (const float*)d_in[9];
    const float* bk      = (const float*)d_in[10];
    const float* wv      = (const float*)d_in[11];
    const float* bv      = (const float*)d_in[12];
    const float* gamma   = (const float*)d_in[13];

    float* ws   = (float*)d_ws;
    float* feat = ws;                        // 16,777,216 f  (x, then feat in place)
    float* Vb   = ws + 16777216;             // 16,777,216 f  (V, later reused as S)
    float* Qb   = ws + 33554432;             //  2,097,152 f
    float* Kb   = ws + 35651584;             //  2,097,152 f
    float* sm   = ws + 37748736;             // small region
    float* bnscale = sm;                     // 256
    float* bnshift = sm + 256;               // 256
    float* Ksum    = sm + 512;               // 128
    float* Vsum    = sm + 640;               // 1024
    float* tl      = sm + 1664;              // 65536
    float* part    = sm + 67200;             // 262144
    float* matT    = sm + 329344;            // 32768

    dim3 blk(256);

    // 1) conv GEMM: x = w_conv @ [s5;s4;s3;s2]   (gather mode, no fcat buffer)
    gemm_f32_wmma<<<dim3(512, 2), blk, 0, stream>>>(
        w_conv, 256, 1024, 0, nullptr, s5, s4, s3, s2, 1, nullptr, 0, feat);

    // 2) BN train-mode stats + fused BN/ReLU in place -> feat
    bn_stats<<<256, blk, 0, stream>>>(feat, bn_g, bn_b, bnscale, bnshift);
    bn_relu_inplace<<<16384, blk, 0, stream>>>(feat, bnscale, bnshift);

    // 3) Q, K, V projections (WMMA GEMM + bias)
    gemm_f32_wmma<<<dim3(512, 1), blk, 0, stream>>>(
        wq, 32, 256, 0, feat, nullptr, nullptr, nullptr, nullptr, 0, bq, 0, Qb);
    gemm_f32_wmma<<<dim3(512, 1), blk, 0, stream>>>(
        wk, 32, 256, 0, feat, nullptr, nullptr, nullptr, nullptr, 0, bk, 0, Kb);
    gemm_f32_wmma<<<dim3(512, 2), blk, 0, stream>>>(
        wv, 256, 256, 0, feat, nullptr, nullptr, nullptr, nullptr, 0, bv, 0, Vb);

    // 4) L2 normalize Q,K per column; row sums; tailor
    l2norm_inplace<<<256, blk, 0, stream>>>(Qb, Kb);
    rowsum16k<<<128, blk, 0, stream>>>(Kb, Ksum, 32);
    rowsum16k<<<1024, blk, 0, stream>>>(Vb, Vsum, 256);
    tailor_kernel<<<256, blk, 0, stream>>>(Qb, Ksum, tl);

    // 5) matrix[b][q][c] = Kn @ V^T  (split-K, deterministic), transposed store
    kv_partial<<<dim3(8, 8, 4), blk, 0, stream>>>(Kb, Vb, part);
    kv_reduceT<<<128, blk, 0, stream>>>(part, matT);

    // 6) S = Vsum + matrix^T @ Qn   (batch-strided A, per-batch bias); S aliases V
    gemm_f32_wmma<<<dim3(512, 2), blk, 0, stream>>>(
        matT, 256, 32, 8192, Qb, nullptr, nullptr, nullptr, nullptr, 0, Vsum, 256, Vb);

    // 7) out = nan_fix(gamma * S * tailor) + feat, remapped to [B,C,H,W]
    final_kernel<<<16384, blk, 0, stream>>>(Vb, tl, feat, gamma, (float*)d_out);
}